// MinGRU_43817256354306
// MI455X (gfx1250) — compile-verified
//
#include <hip/hip_runtime.h>

// MinGRU: k = x@Wz+bz ; h~ = x@Wh+bh ; a = 1-sigmoid(k) ; b = sigmoid(k)*g(h~)
// h[t] = a[t]*h[t-1] + b[t]   (identical to reference's log-space scan)

typedef float v2f __attribute__((ext_vector_type(2)));
typedef float v8f __attribute__((ext_vector_type(8)));
typedef unsigned int u32x4 __attribute__((ext_vector_type(4)));
typedef int i32x4 __attribute__((ext_vector_type(4)));
typedef int i32x8 __attribute__((ext_vector_type(8)));

#define BATCH 8
#define TLEN  4096
#define DIN   512
#define DH    512
#define MM    (BATCH * TLEN)   // 32768 rows
// TDM pads 2 DWORDs after every 256 DWORDs -> effective LDS row stride 516
// (516 % 64 == 4 -> ds reads across lanes m=0..15 hit distinct banks)
#define LDA   516

__global__ __launch_bounds__(256) void mingru_gemm_gates(
    const float* __restrict__ x,
    const float* __restrict__ Wz, const float* __restrict__ bz,
    const float* __restrict__ Wh, const float* __restrict__ bh,
    float* __restrict__ a_out,   // ws  : a = 1 - z
    float* __restrict__ b_out)   // out : b = z * g(h~)
{
    __shared__ float As[16 * LDA];

    const int tid  = threadIdx.x;
    const int wave = tid >> 5;
    const int lane = tid & 31;

    // block -> (16-row M tile, group of 8 N tiles); wave -> one 16x16 tile
    const int mt = blockIdx.x >> 2;
    const int nt = ((blockIdx.x & 3) << 3) + wave;
    const int m0 = mt << 4;
    const int n0 = nt << 4;

    // --- Stage A tile (16 x 512 f32) into LDS via the Tensor Data Mover ----
    // D# per CDNA5 ISA ch.8: 2D tile, data_size=4B, pad 2 DW per 256 DW.
    if (wave == 0) {
        unsigned long long gaddr =
            (unsigned long long)(const void*)(x + (size_t)m0 * DIN);
        unsigned int ldsoff = (unsigned int)(unsigned long long)(void*)As;

        u32x4 g0;
        g0[0] = 1u;                                      // count=1 (valid D#)
        g0[1] = ldsoff;                                  // lds_addr (bytes)
        g0[2] = (unsigned int)(gaddr & 0xFFFFFFFFull);   // global_addr[31:0]
        g0[3] = (unsigned int)(gaddr >> 32) | (2u << 30);// global_addr[56:32] | type=2

        i32x8 g1;
        // data_size=2 (4B) | pad_enable | pad_interval=7 (256 DW) | pad_amount=1 (2 DW)
        g1[0] = (int)((2u << 16) | (1u << 20) | (7u << 22) | (1u << 25));
        g1[1] = (int)(512u << 16);       // ab_addr=0 | tensor_dim0[15:0]=512
        g1[2] = (int)(32768u << 16);     // tensor_dim0[31:16]=0 | tensor_dim1[15:0]=32768
        g1[3] = (int)(512u << 16);       // tensor_dim1[31:16]=0 | tile_dim0=512
        g1[4] = 16;                      // tile_dim1=16 | tile_dim2=0
        g1[5] = 512;                     // tensor_dim0_stride[31:0]=512
        g1[6] = 0;                       // stride0 hi | stride1 lo
        g1[7] = 0;

        i32x4 z4 = {0, 0, 0, 0};         // 2D tensor: groups 2/3 unused
        i32x8 z8 = {0, 0, 0, 0, 0, 0, 0, 0};
        __builtin_amdgcn_tensor_load_to_lds(g0, g1, z4, z4, z8, 0);
        __builtin_amdgcn_s_wait_tensorcnt(0);
    }
    __syncthreads();

    // WMMA f32 16x16x4 lane mapping (ISA 7.12.2, 32-bit A 16x4 / B 4x16 / C 16x16)
    const int n    = lane & 15;        // N within tile (B/C/D)
    const int m    = lane & 15;        // M within tile (A)
    const int half = lane >> 4;        // 0 or 1
    const int kb   = half << 1;        // K sub-offset: 0 or 2

    v8f accz = {};
    v8f acch = {};
    // two K-halves: TDM inserts a 2-DWORD pad after the first 256 DWORDs of each row
    #pragma unroll
    for (int hf = 0; hf < 2; ++hf) {
        const float* pa = &As[m * LDA + kb + hf * 258];
        const float* pz = Wz + (size_t)(hf * 256 + kb) * DH + (n0 + n);
        const float* ph = Wh + (size_t)(hf * 256 + kb) * DH + (n0 + n);
        #pragma unroll 4
        for (int k = 0; k < 256; k += 4) {
            v2f a;  a.x  = pa[0];  a.y  = pa[1];        // ds_load_2addr_b32
            v2f vz; vz.x = pz[0];  vz.y = pz[DH];       // rows kb,kb+1 (coalesced over lanes)
            v2f vh; vh.x = ph[0];  vh.y = ph[DH];
            accz = __builtin_amdgcn_wmma_f32_16x16x4_f32(
                false, a, false, vz, (short)0, accz, false, false);
            acch = __builtin_amdgcn_wmma_f32_16x16x4_f32(
                false, a, false, vh, (short)0, acch, false, false);
            pa += 4;
            pz += (size_t)4 * DH;
            ph += (size_t)4 * DH;
        }
    }

    // epilogue: bias + gates (branchless, fast rcp), write a -> ws, b -> out
    const float bzv = bz[n0 + n];
    const float bhv = bh[n0 + n];
    const int   col = n0 + n;
    const int   rb  = m0 + (half << 3);     // D: lanes 16-31 hold M = r+8
    #pragma unroll
    for (int r = 0; r < 8; ++r) {
        float kv = accz[r] + bzv;
        float hv = acch[r] + bhv;
        float av = __builtin_amdgcn_rcpf(1.0f + __expf(kv));   // sigmoid(-k) = 1-z
        float zv = 1.0f - av;
        float sg = __builtin_amdgcn_rcpf(1.0f + __expf(-hv));  // sigmoid(h~)
        float gv = (hv >= 0.0f) ? (hv + 0.5f) : sg;            // v_cndmask
        size_t idx = (size_t)(rb + r) * DH + col;
        a_out[idx] = av;
        b_out[idx] = zv * gv;
    }
}

// One thread per (batch, channel): sequential recurrence over T, in-place on d_out.
__global__ __launch_bounds__(256) void mingru_scan(
    const float* __restrict__ a_in, float* __restrict__ hb)
{
    const int gid = blockIdx.x * 256 + threadIdx.x;   // 0..4095
    const int b   = gid >> 9;
    const int h   = gid & 511;
    size_t base = (size_t)b * TLEN * DH + h;

    float s = 0.0f;
    for (int t = 0; t < TLEN; t += 4) {
        size_t i0 = base + (size_t)t * DH;
        // batch the independent loads ahead of the dependent FMA chain
        float a0 = a_in[i0];          float b0 = hb[i0];
        float a1 = a_in[i0 +   DH];   float b1 = hb[i0 +   DH];
        float a2 = a_in[i0 + 2*DH];   float b2 = hb[i0 + 2*DH];
        float a3 = a_in[i0 + 3*DH];   float b3 = hb[i0 + 3*DH];
        __builtin_prefetch(&a_in[i0 + 16 * DH], 0, 0);   // global_prefetch_b8
        __builtin_prefetch(&hb[i0 + 16 * DH], 0, 0);
        s = fmaf(a0, s, b0); hb[i0]        = s;
        s = fmaf(a1, s, b1); hb[i0 +   DH] = s;
        s = fmaf(a2, s, b2); hb[i0 + 2*DH] = s;
        s = fmaf(a3, s, b3); hb[i0 + 3*DH] = s;
    }
}

extern "C" void kernel_launch(void* const* d_in, const int* in_sizes, int n_in,
                              void* d_out, int out_size, void* d_ws, size_t ws_size,
                              hipStream_t stream) {
    const float* x  = (const float*)d_in[0];
    const float* Wz = (const float*)d_in[1];
    const float* bz = (const float*)d_in[2];
    const float* Wh = (const float*)d_in[3];
    const float* bh = (const float*)d_in[4];
    float* out = (float*)d_out;          // pass1: b-values, pass2: h (in place)
    float* aws = (float*)d_ws;           // needs MM*DH*4 = 64 MB scratch

    // 2048 M-tiles * 32 N-tiles, 8 wave-tiles per 256-thread block -> 8192 blocks
    mingru_gemm_gates<<<dim3((MM / 16) * 4), dim3(256), 0, stream>>>(
        x, Wz, bz, Wh, bh, aws, out);

    // 4096 channels, one thread each
    mingru_scan<<<dim3((BATCH * DH) / 256), dim3(256), 0, stream>>>(aws, out);
}